// MaskedLinear_6811818132414
// MI455X (gfx1250) — compile-verified
//
#include <hip/hip_runtime.h>
#include <hip/hip_bf16.h>
#include <stdint.h>

// Problem constants (from the reference)
#define LENGTH 4096
#define BATCH  8192
#define KSIZE  7
#define RADIUS 3

#ifndef __has_builtin
#define __has_builtin(x) 0
#endif

#if __has_builtin(__builtin_amdgcn_tensor_load_to_lds)
#define HAVE_TDM 1
#else
#define HAVE_TDM 0
#endif

#if __has_builtin(__builtin_amdgcn_wmma_f32_16x16x4_f32)
#define HAVE_WMMA_F32 1
#else
#define HAVE_WMMA_F32 0
#endif

// therock-10.0 headers (amdgpu-toolchain) ship the TDM header and use the
// 6-arg builtin; ROCm 7.2 clang-22 declares the 5-arg form.
#if __has_include(<hip/amd_detail/amd_gfx1250_TDM.h>)
#define TDM_SIX_ARGS 1
#else
#define TDM_SIX_ARGS 0
#endif

typedef unsigned int u32x4 __attribute__((ext_vector_type(4)));
typedef int          i32x4 __attribute__((ext_vector_type(4)));
typedef int          i32x8 __attribute__((ext_vector_type(8)));
typedef float        f32x4 __attribute__((ext_vector_type(4)));
typedef float        v2f   __attribute__((ext_vector_type(2)));
typedef float        v8f   __attribute__((ext_vector_type(8)));

// ---------------------------------------------------------------------------
// Issue one TDM 2D-tile descriptor: tile_w floats x 7 rows from wband
// (row stride LENGTH floats) -> LDS at lds_off, optionally padding each
// 128-DWORD row by +4 DWORDs (stride 132) to kill bank conflicts.
// Must be called from a single wave (uniform branch); TDM ignores EXEC.
// ---------------------------------------------------------------------------
__device__ __forceinline__ void tdm_stage_band(const float* wband, int obase,
                                               unsigned lds_off, int tile_w,
                                               bool pad132) {
#if HAVE_TDM
    const uint64_t ga = (uint64_t)(uintptr_t)wband + (uint64_t)obase * 4ull;
    u32x4 g0;
    g0[0] = 1u;                                            // count=1, no gather
    g0[1] = lds_off;                                       // lds_addr (bytes)
    g0[2] = (unsigned)(ga & 0xFFFFFFFFull);                // global_addr[31:0]
    g0[3] = (unsigned)((ga >> 32) & 0x1FFFFFFull)          // global_addr[56:32]
          | (2u << 30);                                    // type=2 ("image")

    i32x8 g1;
    unsigned w0 = (2u << 16);                              // data_size=2 (4 bytes)
    if (pad132) w0 |= (1u << 20)                           // pad_enable
               |  (6u << 22)                               // pad_interval: 128 DW
               |  (3u << 25);                              // pad_amount: 4 DW
    g1[0] = (int)w0;
    g1[1] = (int)(((unsigned)LENGTH & 0xFFFFu) << 16);     // tensor_dim0[15:0]
    g1[2] = (int)(((unsigned)LENGTH >> 16)                 // tensor_dim0[31:16]
          | ((unsigned)KSIZE << 16));                      // tensor_dim1[15:0]
    g1[3] = (int)((unsigned)tile_w << 16);                 // tile_dim0
    g1[4] = (int)KSIZE;                                    // tile_dim1 (tile_dim2=0)
    g1[5] = (int)LENGTH;                                   // tensor_dim0_stride[31:0]
    g1[6] = 0;
    g1[7] = 0;

    i32x4 z4 = {0, 0, 0, 0};                               // groups 2/3: 2D tensor
#if TDM_SIX_ARGS
    i32x8 z8 = {0, 0, 0, 0, 0, 0, 0, 0};
    __builtin_amdgcn_tensor_load_to_lds(g0, g1, z4, z4, z8, 0);
#else
    __builtin_amdgcn_tensor_load_to_lds(g0, g1, z4, z4, 0);
#endif
    __builtin_amdgcn_s_wait_tensorcnt(0);                  // TENSORcnt -> 0
#else
    (void)wband; (void)obase; (void)lds_off; (void)tile_w; (void)pad132;
#endif
}

// ---------------------------------------------------------------------------
// Kernel 1: pack the 7 used diagonals of W (4096x4096) into compact planes
//   Wband[d][o] = W[o, o+d-3]  (zero outside [0,LENGTH))
// 112 KB total; lives in d_ws and stays L2-resident for kernel 2.
// ---------------------------------------------------------------------------
__global__ void band_pack_kernel(const float* __restrict__ W,
                                 float* __restrict__ wband) {
    int idx = blockIdx.x * blockDim.x + threadIdx.x;
    if (idx >= KSIZE * LENGTH) return;
    int d = idx >> 12;              // idx / 4096
    int o = idx & (LENGTH - 1);     // idx % 4096
    int i = o + d - RADIUS;
    float v = 0.0f;
    if (i >= 0 && i < LENGTH) v = W[(size_t)o * LENGTH + (size_t)i];
    wband[(size_t)d * LENGTH + (size_t)o] = v;
}

#if HAVE_WMMA_F32
// ===========================================================================
// WMMA path: out-tile (16b x 16o) = sum of 6 chained V_WMMA_F32_16X16X4_F32
// over K-chunks i0 = o0-4+4c. Each wave owns one 16-wide o-block and loops
// over b-tiles, so its 6 B fragments (masked band weights) are built ONCE
// from the TDM-staged LDS tile and stay in registers.
//   A (16x4):  lane L -> M=L%16, v0:K=2*(L/16), v1:K=2*(L/16)+1
//              => one global_load_b64 of x[b0+L%16][i0+2*(L/16)] per chunk.
//   B (4x16):  lane L -> N=L%16, v0:K=2*(L/16), v1:K=2*(L/16)+1
//              B[K][N] = maskedW[o0+N][i0+K] = Wband[d][o0+N], d=4c+K-N-1.
//   D (16x16): VGPR r, lane L -> out[b0 + r + 8*(L/16)][o0 + L%16].
// Out-of-band entries are zero in Wband, so B self-masks the edges; A chunk
// base columns are clamped into [0, LENGTH-4] so edge loads never fault
// (their products are multiplied by zero B).
// ===========================================================================
#define WBLK_OBLKS 8                 // 8 waves -> 8 o-blocks -> 128 columns
#define WBLK_ROWS  128               // b rows per block -> 8 b-tiles per wave
#define PSTRIDE    132               // TDM-padded LDS plane stride (floats)

__global__ __launch_bounds__(256)
void banded_linear_wmma_kernel(const float* __restrict__ x,
                               const float* __restrict__ wband,
                               float* __restrict__ out) {
    __shared__ float wlds[KSIZE * PSTRIDE];   // 7 x 132 floats (~3.6 KB)

    const int obase = blockIdx.x * (WBLK_OBLKS * 16);
    const int t     = (int)threadIdx.x;

    if (t < 32) {                    // wave 0 issues the tensor DMA
        tdm_stage_band(wband, obase, (unsigned)(uintptr_t)(&wlds[0]),
                       WBLK_OBLKS * 16, /*pad132=*/true);
    }
    __syncthreads();

    const int wave = t >> 5;         // 0..7
    const int lane = t & 31;
    const int N    = lane & 15;      // output column within o-block
    const int hi   = lane >> 4;      // lane half -> K pair {2hi, 2hi+1}
    const int o0   = obase + wave * 16;

    // Build the 6 loop-invariant B fragments + clamped A column bases.
    v2f Bfrag[6];
    int colbase[6];
#pragma unroll
    for (int c = 0; c < 6; ++c) {
        const int i0 = o0 - 4 + 4 * c;
        colbase[c] = i0 < 0 ? 0 : (i0 > LENGTH - 4 ? LENGTH - 4 : i0);
        const int K0 = 2 * hi;
        const int d0 = 4 * c + K0 - N - 1;
        const int d1 = d0 + 1;
        float b0v = 0.0f, b1v = 0.0f;
        if (d0 >= 0 && d0 < KSIZE) b0v = wlds[d0 * PSTRIDE + (o0 - obase) + N];
        if (d1 >= 0 && d1 < KSIZE) b1v = wlds[d1 * PSTRIDE + (o0 - obase) + N];
        Bfrag[c][0] = b0v;
        Bfrag[c][1] = b1v;
    }

    const int row0 = (int)blockIdx.y * WBLK_ROWS;

    for (int bt = 0; bt < WBLK_ROWS / 16; ++bt) {
        const int b0 = row0 + bt * 16;
        const float* xb = x + (size_t)(b0 + N) * LENGTH + 2 * hi;

        v8f acc = {0.f, 0.f, 0.f, 0.f, 0.f, 0.f, 0.f, 0.f};
#pragma unroll
        for (int c = 0; c < 6; ++c) {
            v2f a = *(const v2f*)(xb + colbase[c]);        // global_load_b64
            acc = __builtin_amdgcn_wmma_f32_16x16x4_f32(
                false, a, false, Bfrag[c], (short)0, acc, false, false);
        }

        // D: VGPR r, this lane -> out[b0 + r + 8*hi][o0 + N]; 64 B contiguous
        // per 16-lane half per store; out is write-once -> non-temporal.
        float* op = out + (size_t)(b0 + 8 * hi) * LENGTH + (size_t)(o0 + N);
#pragma unroll
        for (int r = 0; r < 8; ++r) {
            __builtin_nontemporal_store(acc[r], op + (size_t)r * LENGTH);
        }
    }
}
#endif  // HAVE_WMMA_F32

// ===========================================================================
// Scalar fallback (used only if the f32 WMMA builtin is unavailable):
// TDM-staged weights -> registers; 3x b128 x-loads, 28 FMAs, 1 NT b128 store
// per thread per row.
// ===========================================================================
#define TILE_O 1024
#define ROWS_PER_BLOCK 16
#define NTHREADS 256

__global__ __launch_bounds__(NTHREADS)
void banded_linear_scalar_kernel(const float* __restrict__ x,
                                 const float* __restrict__ wband,
                                 float* __restrict__ out) {
    __shared__ float wlds[KSIZE * TILE_O];   // 28 KB

    const int obase = blockIdx.x * TILE_O;
    const int t     = (int)threadIdx.x;

#if HAVE_TDM
    if (t < 32) {
        tdm_stage_band(wband, obase, (unsigned)(uintptr_t)(&wlds[0]),
                       TILE_O, /*pad132=*/false);
    }
    __syncthreads();
#else
    for (int i = t; i < KSIZE * TILE_O / 4; i += NTHREADS) {
        int d = i / (TILE_O / 4);
        int c = i % (TILE_O / 4);
        ((f32x4*)wlds)[i] =
            ((const f32x4*)(wband + (size_t)d * LENGTH + obase))[c];
    }
    __syncthreads();
#endif

    float w[KSIZE][4];
#pragma unroll
    for (int d = 0; d < KSIZE; ++d) {
        f32x4 wv = ((const f32x4*)wlds)[d * (TILE_O / 4) + t];
        w[d][0] = wv.x; w[d][1] = wv.y; w[d][2] = wv.z; w[d][3] = wv.w;
    }

    const int  o0       = obase + t * 4;
    const bool at_left  = (o0 == 0);
    const bool at_right = (o0 == LENGTH - 4);
    const int  row0     = (int)blockIdx.y * ROWS_PER_BLOCK;
    const f32x4 zero4   = {0.f, 0.f, 0.f, 0.f};

    for (int rr = 0; rr < ROWS_PER_BLOCK; ++rr) {
        const size_t rbase = (size_t)(row0 + rr) * LENGTH;
        f32x4 va = at_left  ? zero4 : *(const f32x4*)(x + rbase + o0 - 4);
        f32x4 vb = *(const f32x4*)(x + rbase + o0);
        f32x4 vc = at_right ? zero4 : *(const f32x4*)(x + rbase + o0 + 4);

        float r[12] = {va.x, va.y, va.z, va.w,
                       vb.x, vb.y, vb.z, vb.w,
                       vc.x, vc.y, vc.z, vc.w};
        f32x4 acc;
#pragma unroll
        for (int j = 0; j < 4; ++j) {
            float s = 0.0f;
#pragma unroll
            for (int d = 0; d < KSIZE; ++d) s = fmaf(w[d][j], r[1 + j + d], s);
            acc[j] = s;
        }
        __builtin_nontemporal_store(acc, (f32x4*)(out + rbase + o0));
    }
}

// ---------------------------------------------------------------------------
extern "C" void kernel_launch(void* const* d_in, const int* in_sizes, int n_in,
                              void* d_out, int out_size, void* d_ws, size_t ws_size,
                              hipStream_t stream) {
    (void)in_sizes; (void)n_in; (void)out_size; (void)ws_size;
    const float* x  = (const float*)d_in[0];   // (8192, 4096) f32
    const float* W  = (const float*)d_in[1];   // (4096, 4096) f32
    float* out      = (float*)d_out;           // (8192, 4096) f32
    float* wband    = (float*)d_ws;            // 7*4096 floats = 114,688 B scratch

    const int n = KSIZE * LENGTH;
    band_pack_kernel<<<(n + 255) / 256, 256, 0, stream>>>(W, wband);

#if HAVE_WMMA_F32
    dim3 grid(LENGTH / (WBLK_OBLKS * 16), BATCH / WBLK_ROWS);  // (32, 64)
    banded_linear_wmma_kernel<<<grid, 256, 0, stream>>>(x, wband, out);
#else
    dim3 grid(LENGTH / TILE_O, BATCH / ROWS_PER_BLOCK);        // (4, 512)
    banded_linear_scalar_kernel<<<grid, NTHREADS, 0, stream>>>(x, wband, out);
#endif
}